// SHPointCloud_35562329211664
// MI455X (gfx1250) — compile-verified
//
#include <hip/hip_runtime.h>

typedef __attribute__((ext_vector_type(2))) float v2f;
typedef __attribute__((ext_vector_type(8))) float v8f;

#define OUT_DIM 64

// Kernel 1: invert the permutation: inv[ids_3d[n]] = n
__global__ void build_inverse_perm(const int* __restrict__ ids_3d,
                                   int* __restrict__ inv, int n) {
    int i = blockIdx.x * blockDim.x + threadIdx.x;
    if (i < n) inv[ids_3d[i]] = i;
}

// Kernel 2: fully fused SH eval + per-query 16x16 matvec + WMMA MLP (16x16 @ 16x64 + b)
// One wave (32 lanes) handles 16 queries.
__global__ __launch_bounds__(32) void sh_fused(
    const int*   __restrict__ point_ids,   // (M)
    const float* __restrict__ cam_center,  // (3)
    const float* __restrict__ xyz,         // (N,3)
    const float* __restrict__ sh_coeffs,   // (N,16,16)
    const float* __restrict__ confi,       // (N,1,16)
    const float* __restrict__ W,           // (16,64)
    const float* __restrict__ bvec,        // (64)
    const int*   __restrict__ inv,         // (N) inverse permutation
    float*       __restrict__ out,         // [M*64 out_feats][M confi_res]
    int n_query)
{
    __shared__ float shf[16][16];          // sh_feats tile, [query][channel]

    const int lane = threadIdx.x;
    const int q    = lane & 15;
    const int half = lane >> 4;            // 0 or 1
    const int base = blockIdx.x * 16;
    const int m    = base + q;

    const int idx = inv[point_ids[m]];

    // ---- direction & normalization (matches reference eps semantics) ----
    const float dx = cam_center[0] - xyz[idx * 3 + 0];
    const float dy = cam_center[1] - xyz[idx * 3 + 1];
    const float dz = cam_center[2] - xyz[idx * 3 + 2];
    const float nrm = sqrtf(dx * dx + dy * dy + dz * dz);
    const float inr = 1.0f / fmaxf(nrm, 1e-12f);
    const float x = dx * inr, y = dy * inr, z = dz * inr;

    // ---- real SH l=0..3, e3nn order, 'component' normalization ----
    const float SQRT3 = 1.7320508075688772f;
    const float s5    = 2.2360679774997896f;   // sqrt(5)
    const float s7    = 2.6457513110645907f;   // sqrt(7)
    const float c56   = 0.9128709291752769f;   // sqrt(5/6)
    const float c38   = 0.6123724356957945f;   // sqrt(3/8)

    const float y2   = y * y;
    const float x2z2 = x * x + z * z;
    const float sh2_0 = SQRT3 * x * z;
    const float sh2_1 = SQRT3 * x * y;
    const float sh2_2 = y2 - 0.5f * x2z2;
    const float sh2_3 = SQRT3 * y * z;
    const float sh2_4 = 0.5f * SQRT3 * (z * z - x * x);
    const float sh3_0 = c56 * (sh2_0 * z + sh2_4 * x);
    const float sh3_1 = s5 * sh2_0 * y;
    const float sh3_2 = c38 * (4.0f * y2 - x2z2) * x;
    const float sh3_3 = 0.5f * y * (2.0f * y2 - 3.0f * x2z2);
    const float sh3_4 = c38 * z * (4.0f * y2 - x2z2);
    const float sh3_5 = s5 * sh2_4 * y;
    const float sh3_6 = c56 * (sh2_4 * z - sh2_0 * x);

    float Y[16];
    Y[0] = 1.0f;
    Y[1] = SQRT3 * x;  Y[2] = SQRT3 * y;  Y[3] = SQRT3 * z;
    Y[4] = s5 * sh2_0; Y[5] = s5 * sh2_1; Y[6] = s5 * sh2_2;
    Y[7] = s5 * sh2_3; Y[8] = s5 * sh2_4;
    Y[9]  = s7 * sh3_0; Y[10] = s7 * sh3_1; Y[11] = s7 * sh3_2;
    Y[12] = s7 * sh3_3; Y[13] = s7 * sh3_4; Y[14] = s7 * sh3_5;
    Y[15] = s7 * sh3_6;

    // ---- confi_res: dot(confi[idx], Y), lanes 0-15 only (half 1 is duplicate) ----
    if (half == 0) {
        const float4* cp = (const float4*)(confi + (size_t)idx * 16);
        float s = 0.0f;
        #pragma unroll
        for (int d4 = 0; d4 < 4; ++d4) {
            float4 rv = cp[d4];
            s = fmaf(rv.x, Y[d4 * 4 + 0], s);
            s = fmaf(rv.y, Y[d4 * 4 + 1], s);
            s = fmaf(rv.z, Y[d4 * 4 + 2], s);
            s = fmaf(rv.w, Y[d4 * 4 + 3], s);
        }
        out[(size_t)n_query * OUT_DIM + m] = s;
    }

    // ---- sh_feats: per-query 16x16 matvec; this lane does 8 channels ----
    const float* sp = sh_coeffs + (size_t)idx * 256 + (size_t)half * 8 * 16;
    #pragma unroll
    for (int c = 0; c < 8; ++c) {
        const float4* rp = (const float4*)(sp + c * 16);
        float s = 0.0f;
        #pragma unroll
        for (int d4 = 0; d4 < 4; ++d4) {
            float4 rv = rp[d4];
            s = fmaf(rv.x, Y[d4 * 4 + 0], s);
            s = fmaf(rv.y, Y[d4 * 4 + 1], s);
            s = fmaf(rv.z, Y[d4 * 4 + 2], s);
            s = fmaf(rv.w, Y[d4 * 4 + 3], s);
        }
        shf[q][half * 8 + c] = s;
    }
    __syncthreads();   // single-wave WG: compiler fence + S_NOP barrier

    // ---- MLP: out16x64 = shf(16x16) @ W(16x64) + b via V_WMMA_F32_16X16X4_F32 ----
    // f32 16x4 A layout: lane group (half) holds K = 2*half + {0,1} of row M = q.
    const int klo = half * 2;
    v2f afrag[4];
    #pragma unroll
    for (int kk = 0; kk < 4; ++kk) {
        afrag[kk][0] = shf[q][4 * kk + klo + 0];
        afrag[kk][1] = shf[q][4 * kk + klo + 1];
    }

    #pragma unroll
    for (int nt = 0; nt < 4; ++nt) {
        const int col = nt * 16 + q;        // N index for B and C/D
        const float bb = bvec[col];
        v8f acc = {bb, bb, bb, bb, bb, bb, bb, bb};   // bias folded into C

        #pragma unroll
        for (int kk = 0; kk < 4; ++kk) {
            v2f bfrag;                      // B 4x16: VGPR v, group half -> K = 4kk + 2*half + v
            bfrag[0] = W[(4 * kk + klo + 0) * OUT_DIM + col];
            bfrag[1] = W[(4 * kk + klo + 1) * OUT_DIM + col];
            acc = __builtin_amdgcn_wmma_f32_16x16x4_f32(
                false, afrag[kk], false, bfrag, (short)0, acc, false, false);
        }

        // C/D layout: VGPR v holds row M = v + 8*half, col = lane&15
        #pragma unroll
        for (int v = 0; v < 8; ++v) {
            const int row = v + half * 8;
            out[(size_t)(base + row) * OUT_DIM + col] = acc[v];
        }
    }
}

extern "C" void kernel_launch(void* const* d_in, const int* in_sizes, int n_in,
                              void* d_out, int out_size, void* d_ws, size_t ws_size,
                              hipStream_t stream) {
    const int*   point_ids = (const int*)d_in[0];
    const float* cam       = (const float*)d_in[1];
    const int*   ids_3d    = (const int*)d_in[2];
    const float* xyz       = (const float*)d_in[3];
    const float* sh        = (const float*)d_in[4];
    const float* confi     = (const float*)d_in[5];
    const float* W         = (const float*)d_in[6];
    const float* b         = (const float*)d_in[7];
    float* out = (float*)d_out;

    const int n_pts   = in_sizes[2];
    const int n_query = in_sizes[0];

    int* inv = (int*)d_ws;   // n_pts ints of scratch

    build_inverse_perm<<<(n_pts + 255) / 256, 256, 0, stream>>>(ids_3d, inv, n_pts);
    sh_fused<<<n_query / 16, 32, 0, stream>>>(point_ids, cam, xyz, sh, confi,
                                              W, b, inv, out, n_query);
}